// Simplicial_GCN_53601191854811
// MI455X (gfx1250) — compile-verified
//
#include <hip/hip_runtime.h>

#define N    8192
#define HID  30
#define HP   32                 // padded feature stride
#define EDGE_CAP (1u << 20)     // expected ~268K nonzeros; huge safety margin

typedef __attribute__((ext_vector_type(2))) float v2f;
typedef __attribute__((ext_vector_type(8))) float v8f;

// ---------------------------------------------------------------- utilities
__global__ void k_zero_f32(float* p, int n) {
  int i = blockIdx.x * blockDim.x + threadIdx.x;
  if (i < n) p[i] = 0.0f;
}

__global__ void k_init(int* cnt, unsigned* ecnt) {
  int i = blockIdx.x * blockDim.x + threadIdx.x;
  if (i < N) cnt[i] = 0;
  if (i == 0) *ecnt = 0u;
}

// pack 30x30 weight matrices into zero-padded 32x32
__global__ void k_pack_w(const float* __restrict__ W2, const float* __restrict__ W3,
                         float* __restrict__ W2p, float* __restrict__ W3p) {
  int i = blockIdx.x * blockDim.x + threadIdx.x;   // 0..1023
  int r = i >> 5, c = i & 31;
  bool in = (r < HID) & (c < HID);
  W2p[i] = in ? W2[r * HID + c] : 0.0f;
  W3p[i] = in ? W3[r * HID + c] : 0.0f;
}

// ------------------------------------------------- single streaming pass over x
// one block per row: T[row,:] = sum_j x[row,j] * W1[j,:]   (sparse accumulate)
// cnt[j] += 1 for every off-diagonal nonzero (in-degree), edge list append.
__global__ void k_scan(const float* __restrict__ x, const float* __restrict__ W1,
                       float* __restrict__ T, int* __restrict__ cnt,
                       unsigned* __restrict__ ecnt, unsigned* __restrict__ edges) {
  __shared__ float acc[HP];
  const int row = blockIdx.x;
  const int t   = threadIdx.x;                      // 256 threads
  if (t < HP) acc[t] = 0.0f;
  __syncthreads();
  const float* xr = x + (size_t)row * N;
  for (int chunk = 0; chunk < N; chunk += 1024) {   // 1024 floats per block-iter
    const int j0 = chunk + t * 4;
    float4 v = *(const float4*)(xr + j0);
    if (chunk + 1024 < N) __builtin_prefetch(xr + j0 + 1024, 0, 0);
#pragma unroll
    for (int q = 0; q < 4; ++q) {
      float xv = (&v.x)[q];
      if (xv != 0.0f) {
        int j = j0 + q;
        for (int f = 0; f < HID; ++f) atomicAdd(&acc[f], xv * W1[j * HID + f]);
        if (j != row) {                             // self-loops handled analytically
          atomicAdd(&cnt[j], 1);
          unsigned pos = atomicAdd(ecnt, 1u);
          if (pos < EDGE_CAP) edges[pos] = ((unsigned)row << 13) | (unsigned)j;
        }
      }
    }
  }
  __syncthreads();
  if (t < HP) T[(size_t)row * HP + t] = (t < HID) ? acc[t] : 0.0f;
}

__global__ void k_dinv(const int* __restrict__ cnt, float* __restrict__ dinv) {
  int i = blockIdx.x * blockDim.x + threadIdx.x;
  if (i < N) dinv[i] = rsqrtf((float)(cnt[i] + 1));  // +1 = forced self-loop
}

// ------------------------------------------- edge-parallel aggregation (scatter)
// Hacc[dst,:] += dinv[src] * T[src,:]   for every off-diagonal edge
__global__ void k_agg(const unsigned* __restrict__ ecnt, const unsigned* __restrict__ edges,
                      const float* __restrict__ dinv, const float* __restrict__ T,
                      float* __restrict__ Hacc) {
  unsigned nE = *ecnt; if (nE > EDGE_CAP) nE = EDGE_CAP;
  const unsigned wave   = (blockIdx.x * blockDim.x + threadIdx.x) >> 5;
  const unsigned lane   = threadIdx.x & 31u;
  const unsigned nWaves = (gridDim.x * blockDim.x) >> 5;
  for (unsigned e = wave; e < nE; e += nWaves) {
    unsigned w = edges[e];
    unsigned src = w >> 13, dst = w & 8191u;
    if (lane < HID)
      atomicAdd(&Hacc[(size_t)dst * HP + lane], dinv[src] * T[(size_t)src * HP + lane]);
  }
}

// out[i,f] = act( dinv[i] * (Hacc[i,f] + dinv[i]*T[i,f]) + b[f] ),  pad cols -> 0
__global__ void k_fin(const float* __restrict__ Hacc, const float* __restrict__ T,
                      const float* __restrict__ dinv, const float* __restrict__ b,
                      float* __restrict__ Hout, int relu) {
  int idx = blockIdx.x * blockDim.x + threadIdx.x;   // N*HP
  int i = idx >> 5, f = idx & 31;
  float o = 0.0f;
  if (f < HID) {
    float di = dinv[i];
    o = di * (Hacc[idx] + di * T[idx]) + b[f];
    if (relu && o < 0.0f) o = 0.0f;
  }
  Hout[idx] = o;
}

// ----------------------------------- dense 8192x32 @ 32x32 via f32 WMMA (16x16x4)
// one wave per 16-row M-tile; two 16-wide N-tiles; 8 k-steps of V_WMMA_F32_16X16X4_F32
__global__ void k_gemm_wmma(const float* __restrict__ H, const float* __restrict__ Wp,
                            float* __restrict__ T) {
  const int lane = threadIdx.x;        // 32 threads = 1 wave
  const int m0   = blockIdx.x * 16;
  const int half = lane >> 4;          // 0 | 1
  const int ml   = lane & 15;
  v8f c0 = {}; v8f c1 = {};
#pragma unroll
  for (int kk = 0; kk < 8; ++kk) {
    const int kb = kk * 4 + half * 2;  // A: lane holds K = v + 2*(lane/16)
    v2f a, b0, b1;
    a.x  = H[(m0 + ml) * HP + kb];
    a.y  = H[(m0 + ml) * HP + kb + 1];
    b0.x = Wp[kb * HP + ml];           // B: lane holds N = lane%16, K = v + 2*(lane/16)
    b0.y = Wp[(kb + 1) * HP + ml];
    b1.x = Wp[kb * HP + 16 + ml];
    b1.y = Wp[(kb + 1) * HP + 16 + ml];
    c0 = __builtin_amdgcn_wmma_f32_16x16x4_f32(false, a, false, b0, (short)0, c0, false, false);
    c1 = __builtin_amdgcn_wmma_f32_16x16x4_f32(false, a, false, b1, (short)0, c1, false, false);
  }
#pragma unroll
  for (int v = 0; v < 8; ++v) {        // D: lane holds N = lane%16, M = v + 8*(lane/16)
    int m = m0 + v + 8 * half;
    T[m * HP + ml]      = c0[v];
    T[m * HP + 16 + ml] = c1[v];
  }
}

// ------------------------------------------- mean-pool + final linear -> scalar
__global__ void k_pool(const float* __restrict__ H, const float* __restrict__ Wl,
                       const float* __restrict__ bl, float* __restrict__ out) {
  __shared__ float part[1024];
  const int t = threadIdx.x;           // 1024 threads
  const int f = t & 31;
  float s = 0.0f;
  for (int i = (t >> 5); i < N; i += 32) s += H[(size_t)i * HP + f];
  part[t] = s;
  __syncthreads();
  if (t < 32) {
    float tot = 0.0f;
    for (int k = 0; k < 32; ++k) tot += part[t + 32 * k];
    part[t] = tot;
  }
  __syncthreads();
  if (t == 0) {
    float o = 0.0f;
    for (int f2 = 0; f2 < HID; ++f2) o += (part[f2] / (float)N) * Wl[f2];
    out[0] = o + bl[0];
  }
}

// ---------------------------------------------------------------------- driver
extern "C" void kernel_launch(void* const* d_in, const int* in_sizes, int n_in,
                              void* d_out, int out_size, void* d_ws, size_t ws_size,
                              hipStream_t stream) {
  (void)in_sizes; (void)n_in; (void)out_size; (void)ws_size;
  const float* x  = (const float*)d_in[0];
  const float* W1 = (const float*)d_in[1];
  const float* b1 = (const float*)d_in[2];
  const float* W2 = (const float*)d_in[3];
  const float* b2 = (const float*)d_in[4];
  const float* W3 = (const float*)d_in[5];
  const float* b3 = (const float*)d_in[6];
  const float* Wl = (const float*)d_in[7];
  const float* bl = (const float*)d_in[8];
  float* out = (float*)d_out;

  char* p = (char*)d_ws;
  auto take = [&](size_t bytes) { char* r = p; p += (bytes + 255) & ~(size_t)255; return r; };
  int*      cnt   = (int*)take((size_t)N * 4);
  float*    dinv  = (float*)take((size_t)N * 4);
  unsigned* ecnt  = (unsigned*)take(4);
  float*    W2p   = (float*)take((size_t)HP * HP * 4);
  float*    W3p   = (float*)take((size_t)HP * HP * 4);
  float*    T     = (float*)take((size_t)N * HP * 4);
  float*    H     = (float*)take((size_t)N * HP * 4);
  float*    Hacc  = (float*)take((size_t)N * HP * 4);
  unsigned* edges = (unsigned*)take((size_t)EDGE_CAP * 4);

  k_init  <<<N / 256, 256, 0, stream>>>(cnt, ecnt);
  k_pack_w<<<4, 256, 0, stream>>>(W2, W3, W2p, W3p);

  // single pass over x (the 256 MB roofline term): T = x@W1, degrees, edge list
  k_scan<<<N, 256, 0, stream>>>(x, W1, T, cnt, ecnt, edges);
  k_dinv<<<N / 256, 256, 0, stream>>>(cnt, dinv);

  // layer 1: aggregate + bias + relu
  k_zero_f32<<<(N * HP) / 256, 256, 0, stream>>>(Hacc, N * HP);
  k_agg<<<2048, 256, 0, stream>>>(ecnt, edges, dinv, T, Hacc);
  k_fin<<<(N * HP) / 256, 256, 0, stream>>>(Hacc, T, dinv, b1, H, 1);

  // layer 2: WMMA GEMM, aggregate, relu
  k_gemm_wmma<<<N / 16, 32, 0, stream>>>(H, W2p, T);
  k_zero_f32<<<(N * HP) / 256, 256, 0, stream>>>(Hacc, N * HP);
  k_agg<<<2048, 256, 0, stream>>>(ecnt, edges, dinv, T, Hacc);
  k_fin<<<(N * HP) / 256, 256, 0, stream>>>(Hacc, T, dinv, b2, H, 1);

  // layer 3: WMMA GEMM, aggregate, no relu
  k_gemm_wmma<<<N / 16, 32, 0, stream>>>(H, W3p, T);
  k_zero_f32<<<(N * HP) / 256, 256, 0, stream>>>(Hacc, N * HP);
  k_agg<<<2048, 256, 0, stream>>>(ecnt, edges, dinv, T, Hacc);
  k_fin<<<(N * HP) / 256, 256, 0, stream>>>(Hacc, T, dinv, b3, H, 0);

  // mean pool + linear head
  k_pool<<<1, 1024, 0, stream>>>(H, Wl, bl, out);
}